// TextLoss_88734024335439
// MI455X (gfx1250) — compile-verified
//
#include <hip/hip_runtime.h>
#include <stdint.h>

// Problem constants (from reference): BS=32, H=W=640
#define HW_PIX   409600       // 640*640
#define NROWS    13107200     // 32*640*640
#define NB0      2048         // level-0/1 buckets (11 bits)
#define NB2      1024         // level-2 buckets (10 bits)

typedef unsigned int u32;
typedef __attribute__((ext_vector_type(2))) float v2f;
typedef __attribute__((ext_vector_type(8))) float v8f;

struct TLState {
  float posCnt;     // # positive rows (exact, < 2^24)
  float negCnt;     // # negative rows
  float lossPos;    // sum rl*w over positives
  float wPos;       // sum w over positives
  float negTot;     // sum rl over all negatives
  u32   done;       // selection finished
  u32   prefix;     // accumulated radix prefix (bucket path)
  u32   remaining;  // remaining top-k count
  float sumTop;     // accumulated top-k sum
  float kf;         // k as float (for denominator)
};

// ---------------------------------------------------------------------------
// Wave32 sum reduction through the FP32 WMMA pipe (ones-matrix trick).
// A(16x4): lane L vgpr0 -> (M = L%16, K = 2*(L/16)); we put v in vgpr0, 0 in
// vgpr1, B = all ones => D[m,n] = v[m] + v[m+16] (exact f32 FMA, RNE).
// Lane n holds D[0..7,n] in c[0..7]; lane n^16 holds D[8..15,n].
// sum(c) + shfl_xor(.,16) == full 32-lane sum, identical on every lane.
// ---------------------------------------------------------------------------
__device__ __forceinline__ float wave_sum32(float v) {
  v2f a; a[0] = v;    a[1] = 0.0f;
  v2f b; b[0] = 1.0f; b[1] = 1.0f;
  v8f c = {};
  c = __builtin_amdgcn_wmma_f32_16x16x4_f32(false, a, false, b,
                                            (short)0, c, false, false);
  float s = c[0] + c[1] + c[2] + c[3] + c[4] + c[5] + c[6] + c[7];
  s += __shfl_xor(s, 16, 32);
  return s;
}

// ---------------------------------------------------------------------------
// Zero state + all histograms (first 64 KB of workspace). Runs every call so
// graph replays are deterministic.
// ---------------------------------------------------------------------------
__global__ void tl_init(u32* ws) {
  int i = blockIdx.x * blockDim.x + threadIdx.x;
  if (i < 16384) ws[i] = 0u;   // 64 KB
}

// ---------------------------------------------------------------------------
// Pass 1: stream predict/vec_mask/weight once. Emit monotonic u32 key per row
// (0 = positive row, else float_bits(rl)+1), accumulate scalar reductions via
// WMMA wave sums, and build the level-0 (top 11 bits) count+sum histogram.
// ---------------------------------------------------------------------------
__global__ void __launch_bounds__(256)
tl_pass1(const float* __restrict__ pred, const float* __restrict__ vmask,
         const float* __restrict__ wgt, u32* __restrict__ keys,
         TLState* st, u32* gc, float* gs) {
  __shared__ u32   hc[NB0];
  __shared__ float hs[NB0];
  __shared__ float wred[8][8];

  for (int j = threadIdx.x; j < NB0; j += 256) { hc[j] = 0u; hs[j] = 0.0f; }
  __syncthreads();

  float cPos = 0.f, cNeg = 0.f, lp = 0.f, wp = 0.f, nt = 0.f;
  const int stride = gridDim.x * 256;
  for (int i = blockIdx.x * 256 + threadIdx.x; i < NROWS; i += stride) {
    float w  = wgt[i];
    int   bb = i / HW_PIX;
    int   x  = i - bb * HW_PIX;
    int   base = bb * (2 * HW_PIX) + x;
    float p0 = pred[base];
    float p1 = pred[base + HW_PIX];
    float t0 = vmask[base];
    float t1 = vmask[base + HW_PIX];
    float d0 = p0 - t0, d1 = p1 - t1;
    float rl = fmaf(d0, d0, d1 * d1);
    if (w > 0.0f) {
      cPos += 1.0f; lp = fmaf(rl, w, lp); wp += w;
      keys[i] = 0u;
    } else {
      cNeg += 1.0f; nt += rl;
      u32 kb = __float_as_uint(rl) + 1u;   // monotonic key, >=1
      keys[i] = kb;
      atomicAdd(&hc[kb >> 21], 1u);
      atomicAdd(&hs[kb >> 21], rl);
    }
  }

  // Wave reductions through the matrix pipe (EXEC all-1s here: no divergence).
  float r0 = wave_sum32(cPos);
  float r1 = wave_sum32(cNeg);
  float r2 = wave_sum32(lp);
  float r3 = wave_sum32(wp);
  float r4 = wave_sum32(nt);
  int wid = threadIdx.x >> 5, lane = threadIdx.x & 31;
  if (lane == 0) {
    wred[wid][0] = r0; wred[wid][1] = r1; wred[wid][2] = r2;
    wred[wid][3] = r3; wred[wid][4] = r4;
  }
  __syncthreads();
  if (threadIdx.x == 0) {
    float a0 = 0, a1 = 0, a2 = 0, a3 = 0, a4 = 0;
    for (int w = 0; w < 8; ++w) {
      a0 += wred[w][0]; a1 += wred[w][1]; a2 += wred[w][2];
      a3 += wred[w][3]; a4 += wred[w][4];
    }
    atomicAdd(&st->posCnt,  a0);
    atomicAdd(&st->negCnt,  a1);
    atomicAdd(&st->lossPos, a2);
    atomicAdd(&st->wPos,    a3);
    atomicAdd(&st->negTot,  a4);
  }

  for (int j = threadIdx.x; j < NB0; j += 256) {
    u32 c = hc[j];
    if (c) { atomicAdd(&gc[j], c); atomicAdd(&gs[j], hs[j]); }
  }
}

// ---------------------------------------------------------------------------
// Single-block scan over a histogram level: walk buckets from the top, take
// whole buckets while count <= remaining; on the threshold bucket descend
// (levels 0/1) or resolve exactly (level 2: all keys in bucket identical).
// ---------------------------------------------------------------------------
__global__ void tl_scan(TLState* st, const u32* cnt, const float* sum,
                        int nb, int level) {
  __shared__ u32   sc[NB0];
  __shared__ float ss[NB0];

  if (level == 0 && threadIdx.x == 0) {
    unsigned long long pos = (unsigned long long)(st->posCnt + 0.5f);
    unsigned long long neg = (unsigned long long)(st->negCnt + 0.5f);
    unsigned long long k = 3ull * pos; if (k > neg) k = neg;
    st->kf = (float)k;
    if (k == 0ull)      { st->sumTop = 0.0f;      st->done = 1u; }
    else if (k == neg)  { st->sumTop = st->negTot; st->done = 1u; }
    else { st->remaining = (u32)k; st->prefix = 0u; st->sumTop = 0.0f; st->done = 0u; }
  }
  __syncthreads();
  if (st->done) return;

  for (int j = threadIdx.x; j < nb; j += blockDim.x) { sc[j] = cnt[j]; ss[j] = sum[j]; }
  __syncthreads();

  if (threadIdx.x == 0) {
    u32 r = st->remaining;
    float top = st->sumTop;
    u32 pre = st->prefix;
    for (int b = nb - 1; b >= 0; --b) {
      u32 c = sc[b];
      if (c == 0u) continue;
      if (r > c)      { top += ss[b]; r -= c; }
      else if (r == c){ top += ss[b]; r = 0u; st->done = 1u; break; }
      else {
        if (level == 2) {
          u32 key = (pre << 10) | (u32)b;            // exact 32-bit key
          float v = __uint_as_float(key - 1u);
          top += (float)r * v;
          st->done = 1u;
        } else {
          pre = (pre << 11) | (u32)b;                // descend
        }
        break;
      }
    }
    st->remaining = r; st->sumTop = top; st->prefix = pre;
  }
}

// ---------------------------------------------------------------------------
// Refinement histogram over keys matching the current prefix.
// level 1: match key[31:21], bucket = key[20:10]; level 2: match key[31:10],
// bucket = key[9:0]. Values reconstructed from the key (v = asfloat(key-1)).
// ---------------------------------------------------------------------------
__global__ void __launch_bounds__(256)
tl_refine(const u32* __restrict__ keys, const TLState* st,
          u32* gc, float* gs, int level) {
  __shared__ u32   hc[NB0];
  __shared__ float hs[NB0];
  if (st->done) return;                       // uniform across the grid
  const int nb = (level == 1) ? NB0 : NB2;
  for (int j = threadIdx.x; j < nb; j += 256) { hc[j] = 0u; hs[j] = 0.0f; }
  __syncthreads();

  const u32 target = st->prefix;
  const int stride = gridDim.x * 256;
  for (int i = blockIdx.x * 256 + threadIdx.x; i < NROWS; i += stride) {
    u32 k = keys[i];
    if (k == 0u) continue;
    if (level == 1) {
      if ((k >> 21) == target) {
        u32 b = (k >> 10) & 0x7FFu;
        atomicAdd(&hc[b], 1u);
        atomicAdd(&hs[b], __uint_as_float(k - 1u));
      }
    } else {
      if ((k >> 10) == target) {
        u32 b = k & 0x3FFu;
        atomicAdd(&hc[b], 1u);
        atomicAdd(&hs[b], __uint_as_float(k - 1u));
      }
    }
  }
  __syncthreads();
  for (int j = threadIdx.x; j < nb; j += 256) {
    u32 c = hc[j];
    if (c) { atomicAdd(&gc[j], c); atomicAdd(&gs[j], hs[j]); }
  }
}

__global__ void tl_final(const TLState* st, float* out) {
  float denom = 2.0f * st->wPos + 2.0f * st->kf;
  out[0] = (st->lossPos + st->sumTop) / denom / 32.0f;
}

// ---------------------------------------------------------------------------
extern "C" void kernel_launch(void* const* d_in, const int* in_sizes, int n_in,
                              void* d_out, int out_size, void* d_ws, size_t ws_size,
                              hipStream_t stream) {
  const float* pred  = (const float*)d_in[0];
  const float* vmask = (const float*)d_in[1];
  const float* wgt   = (const float*)d_in[2];
  float* out = (float*)d_out;

  char* ws = (char*)d_ws;
  TLState* st = (TLState*)ws;
  u32*   h0c = (u32*)  (ws + 1024);
  float* h0s = (float*)(ws + 1024 + 8192);
  u32*   h1c = (u32*)  (ws + 1024 + 16384);
  float* h1s = (float*)(ws + 1024 + 24576);
  u32*   h2c = (u32*)  (ws + 1024 + 32768);
  float* h2s = (float*)(ws + 1024 + 36864);
  u32*   keys = (u32*) (ws + 65536);          // NROWS u32 = 52.4 MB

  tl_init  <<<64,   256, 0, stream>>>((u32*)ws);
  tl_pass1 <<<2048, 256, 0, stream>>>(pred, vmask, wgt, keys, st, h0c, h0s);
  tl_scan  <<<1,    256, 0, stream>>>(st, h0c, h0s, NB0, 0);
  tl_refine<<<2048, 256, 0, stream>>>(keys, st, h1c, h1s, 1);
  tl_scan  <<<1,    256, 0, stream>>>(st, h1c, h1s, NB0, 1);
  tl_refine<<<2048, 256, 0, stream>>>(keys, st, h2c, h2s, 2);
  tl_scan  <<<1,    256, 0, stream>>>(st, h2c, h2s, NB2, 2);
  tl_final <<<1,    1,   0, stream>>>(st, out);
}